// CliquePotentialsCRF_15470472200684
// MI455X (gfx1250) — compile-verified
//
#include <hip/hip_runtime.h>
#include <stdint.h>

// CliquePotentialsCRF on MI455X (gfx1250):
// energy_pool(G) == M*G*M - (diag(M) outer diag(M)) .* G  with M = P^T P banded (+/-2).
// Persistent kernel: 1 workgroup per (b,c) plane, plane state fully LDS-resident,
// both 128x128 matmuls via V_WMMA_F32_16X16X4_F32 over the banded K range only,
// beta plane staged into padded LDS rows by the Tensor Data Mover (one descriptor),
// per-iteration grid barrier via global atomics for the FW duality gap / done flag.

typedef float v2f __attribute__((ext_vector_type(2)));
typedef float v8f __attribute__((ext_vector_type(8)));

#define LDW        132              // LDS row stride in floats (132 = 128 + 4 TDM pad dwords)
#define PLANES     84               // B*C = 4*21
#define TPB        256              // 8 waves of 32
#define MAX_IT     100
#define TOL_F      1e-3f
#define SMEM_FLOATS (4 * 128 * LDW + 288)   // sM, sB, sV, sT planes + reduction area
#define SMEM_BYTES  (SMEM_FLOATS * 4)       // ~265 KB of the 320 KB WGP LDS

#if __has_builtin(__builtin_amdgcn_tensor_load_to_lds)
#define USE_TDM 1
typedef unsigned int u32x4 __attribute__((ext_vector_type(4)));
typedef int          i32x4 __attribute__((ext_vector_type(4)));
typedef int          i32x8 __attribute__((ext_vector_type(8)));
#else
#define USE_TDM 0
#endif

__device__ __forceinline__ float sigm(float x) {
  // hardware v_exp_f32 + v_rcp_f32 (~1 ulp) instead of IEEE divide expansion
  return __builtin_amdgcn_rcpf(1.0f + __expf(-x));
}

// number of valid 3-cliques covering coordinate y (H = W = 128) == M[y][y]
__device__ __forceinline__ float cliqueCnt(int y) {
  int hi = y < 125 ? y : 125;
  int lo = y - 2 > 0 ? y - 2 : 0;
  return (float)(hi - lo + 1);
}

__global__ void crf_init_ws(int* __restrict__ ctr) {
  if (threadIdx.x < MAX_IT) ctr[threadIdx.x] = 0;
}

__global__ __launch_bounds__(TPB, 1)
void crf_fw_kernel(const float* __restrict__ gBeta, float* __restrict__ gOut,
                   int* __restrict__ gCtr, float* __restrict__ gGap) {
  extern __shared__ float smem[];
  float* sM = smem;                  // M matrix (constant, 0..3 integer values)
  float* sB = smem + 1 * 128 * LDW;  // beta plane (TDM destination, padded rows)
  float* sV = smem + 2 * 128 * LDW;  // v plane
  float* sT = smem + 3 * 128 * LDW;  // T1 = G*M intermediate
  float* sR = smem + 4 * 128 * LDW;  // 256-float reduction area

  const int tid  = threadIdx.x;
  const int lane = tid & 31;
  const int wv   = tid >> 5;               // wave id 0..7 -> 16-row strip
  const int wg   = blockIdx.x;             // plane id 0..83
  const long planeOff = (long)wg * (128 * 128);

#if USE_TDM
  // ---- Tensor Data Mover: stage the 128x128 f32 beta plane into sB with
  //      4-dword padding after every 128 dwords (-> LDS row stride 132). ----
  if (wv == 0) {                           // wave-uniform; TDM ignores EXEC
    unsigned long long gaddr = (unsigned long long)(uintptr_t)(gBeta + planeOff);
    unsigned int       laddr = (unsigned int)(uintptr_t)sB;   // flat addr[31:0] == LDS offset
    u32x4 g0;
    g0[0] = 1u;                                            // count=1, user descriptor
    g0[1] = laddr;                                         // lds_addr (bytes)
    g0[2] = (unsigned int)(gaddr & 0xffffffffu);           // global_addr[31:0]
    g0[3] = (unsigned int)((gaddr >> 32) & 0x01ffffffu)    // global_addr[56:32]
          | (2u << 30);                                    // D# type = 2 ("image")
    i32x8 g1;
    g1[0] = (2 << 16)   // data_size = 4 bytes
          | (1 << 20)   // pad_enable
          | (6 << 22)   // pad_interval: 2^6 * 8B = 128 dwords
          | (3 << 25);  // pad_amount: 3+1 = 4 dwords
    g1[1] = (128 << 16);            // tensor_dim0[15:0] = 128 (bits 63:48)
    g1[2] = (128 << 16);            // tensor_dim1[15:0] = 128 (bits 95:80)
    g1[3] = (128 << 16);            // tile_dim0 = 128 (bits 127:112)
    g1[4] = 128;                    // tile_dim1 = 128 (bits 143:128)
    g1[5] = 128;                    // tensor_dim0_stride = 128 (bits 191:160)
    g1[6] = 0;
    g1[7] = 0;
    i32x4 gz = {0, 0, 0, 0};
#if __has_include(<hip/amd_detail/amd_gfx1250_TDM.h>)
    i32x8 gz8 = {0, 0, 0, 0, 0, 0, 0, 0};
    __builtin_amdgcn_tensor_load_to_lds(g0, g1, gz, gz, gz8, 0);   // clang-23 6-arg form
#else
    __builtin_amdgcn_tensor_load_to_lds(g0, g1, gz, gz, 0);        // ROCm 7.2 5-arg form
#endif
  }
#endif

  // ---- one-time plane setup: M (procedural), v = 0 (beta via TDM above) ----
  for (int k = tid; k < 128 * 128; k += TPB) {
    int i = k >> 7, j = k & 127;
    int mn = i < j ? i : j, mx = i < j ? j : i;
    int lo = mx - 2 > 0 ? mx - 2 : 0;
    int hi = mn < 125 ? mn : 125;
    int c  = hi - lo + 1; if (c < 0) c = 0;
    sM[i * LDW + j] = (float)c;
    sV[i * LDW + j] = 0.0f;
#if !USE_TDM
    sB[i * LDW + j] = gBeta[planeOff + k];
#endif
  }
#if USE_TDM
  if (wv == 0) __builtin_amdgcn_s_wait_tensorcnt(0);
#endif
  __syncthreads();

  // WMMA f32 16x16x4 lane mapping
  const int r0   = wv << 4;                // strip base row
  const int arow = r0 + (lane & 15);       // A: lane -> M row
  const int koff = (lane >> 4) << 1;       // K half select (0 or 2)
  const int ncol = lane & 15;              // B/C/D: lane -> N col
  const int drow = (lane >> 4) << 3;       // C/D: row offset 0 or 8

  float sreg[64];                          // this thread's s_t strip values

  for (int t = 0; t < MAX_IT; ++t) {
    // ================= T1 = G * M  (G = sigmoid(v+beta), M banded +/-2) ======
    v2f a[32];                             // full-K A fragments of G strip
#pragma unroll
    for (int ks = 0; ks < 32; ++ks) {
      int k = (ks << 2) + koff;
      a[ks].x = sigm(sV[arow * LDW + k]     + sB[arow * LDW + k]);
      a[ks].y = sigm(sV[arow * LDW + k + 1] + sB[arow * LDW + k + 1]);
    }
#pragma unroll
    for (int nt = 0; nt < 8; ++nt) {
      const int n0 = nt << 4;
      v8f c = {0.f, 0.f, 0.f, 0.f, 0.f, 0.f, 0.f, 0.f};
#pragma unroll
      for (int q = 0; q < 6; ++q) {        // band: ks in [4nt-1, 4nt+4]
        const int ks = (nt << 2) - 1 + q;  // compile-time after unroll
        if (ks >= 0 && ks <= 31) {
          const int kb = (ks << 2) + koff;
          v2f b;
          b.x = sM[kb * LDW + n0 + ncol];
          b.y = sM[(kb + 1) * LDW + n0 + ncol];
          c = __builtin_amdgcn_wmma_f32_16x16x4_f32(
                  false, a[ks], false, b, (short)0, c, false, false);
        }
      }
#pragma unroll
      for (int r = 0; r < 8; ++r)
        sT[(r0 + r + drow) * LDW + n0 + ncol] = c[r];
    }
    __syncthreads();

    // ================= T2 = M * T1, fused: s = cnt*g - T2, gap += g*(v-s) ====
    const int kstart = (wv << 2) - 1;      // band: ks in [4wv-1, 4wv+4] (wave-uniform)
    v2f a2[6];
#pragma unroll
    for (int q = 0; q < 6; ++q) {
      const int ks = kstart + q;
      if (ks >= 0 && ks <= 31) {
        const int k = (ks << 2) + koff;
        a2[q].x = sM[arow * LDW + k];
        a2[q].y = sM[arow * LDW + k + 1];
      } else {
        a2[q].x = 0.f; a2[q].y = 0.f;
      }
    }
    float gapAcc = 0.f;
#pragma unroll
    for (int nt = 0; nt < 8; ++nt) {
      const int n0 = nt << 4;
      v8f c = {0.f, 0.f, 0.f, 0.f, 0.f, 0.f, 0.f, 0.f};
#pragma unroll
      for (int q = 0; q < 6; ++q) {
        const int ks = kstart + q;
        if (ks >= 0 && ks <= 31) {         // wave-uniform branch, EXEC stays full
          const int kb = (ks << 2) + koff;
          v2f b;
          b.x = sT[kb * LDW + n0 + ncol];
          b.y = sT[(kb + 1) * LDW + n0 + ncol];
          c = __builtin_amdgcn_wmma_f32_16x16x4_f32(
                  false, a2[q], false, b, (short)0, c, false, false);
        }
      }
#pragma unroll
      for (int r = 0; r < 8; ++r) {        // elementwise on D tile, in registers
        const int i = r0 + r + drow;
        const int n = n0 + ncol;
        const float vv = sV[i * LDW + n];
        const float g  = sigm(vv + sB[i * LDW + n]);
        const float s  = cliqueCnt(i) * cliqueCnt(n) * g - c[r]; // -(back - cnt*g), ALPHA=1
        gapAcc += g * (vv - s);
        sreg[(nt << 3) + r] = s;
      }
    }

    // ================= block gap reduction ===================================
    sR[tid] = gapAcc;
    __syncthreads();
#pragma unroll
    for (int off = 128; off >= 1; off >>= 1) {
      if (tid < off) sR[tid] += sR[tid + off];
      __syncthreads();
    }

    // ================= grid barrier + deterministic global gap sum ==========
    if (tid == 0) {
      volatile float* vGap = gGap + (size_t)t * 96;
      vGap[wg] = sR[0];
      __threadfence();
      atomicAdd(&gCtr[t], 1);
      while (((volatile int*)gCtr)[t] < PLANES)
        __builtin_amdgcn_s_sleep(1);
      __threadfence();
      float tot = 0.f;
      for (int w = 0; w < PLANES; ++w)     // fixed order -> deterministic
        tot += vGap[w];
      sR[0] = tot;
    }
    __syncthreads();
    const float gap = sR[0];               // identical across whole grid
    if (gap < TOL_F) break;                // reference freezes v -> early exit is exact

    // ================= v update (each element owned by exactly one thread) ===
    const float gamma = 2.0f / ((float)t + 2.0f);
#pragma unroll
    for (int nt = 0; nt < 8; ++nt) {
#pragma unroll
      for (int r = 0; r < 8; ++r) {
        const int i = r0 + r + drow;
        const int n = (nt << 4) + ncol;
        const float vv = sV[i * LDW + n];
        sV[i * LDW + n] = vv + gamma * (sreg[(nt << 3) + r] - vv);
      }
    }
    __syncthreads();
  }

  // ---- output: -(beta + v), coalesced ----
  __syncthreads();
  for (int k = tid; k < 128 * 128; k += TPB) {
    int i = k >> 7, j = k & 127;
    gOut[planeOff + k] = -(sB[i * LDW + j] + sV[i * LDW + j]);
  }
}

extern "C" void kernel_launch(void* const* d_in, const int* in_sizes, int n_in,
                              void* d_out, int out_size, void* d_ws, size_t ws_size,
                              hipStream_t stream) {
  (void)in_sizes; (void)n_in; (void)out_size; (void)ws_size;
  const float* beta = (const float*)d_in[0];
  float* out = (float*)d_out;
  int*   ctr = (int*)d_ws;                              // MAX_IT counters
  float* gap = (float*)((char*)d_ws + 512);             // MAX_IT * 96 gap slots

  (void)hipFuncSetAttribute(reinterpret_cast<const void*>(crf_fw_kernel),
                            hipFuncAttributeMaxDynamicSharedMemorySize, SMEM_BYTES);

  crf_init_ws<<<1, 128, 0, stream>>>(ctr);
  crf_fw_kernel<<<PLANES, TPB, SMEM_BYTES, stream>>>(beta, out, ctr, gap);
}